// IHAConv_82102594830700
// MI455X (gfx1250) — compile-verified
//
#include <hip/hip_runtime.h>
#include <hip/hip_bf16.h>

#define N_NODES 50000
#define N_EDGES 800000
#define IN_FEATS 256
#define NUM_HEADS 8
#define OUT_FEATS 32
#define HD 256   // NUM_HEADS*OUT_FEATS
#define NEG_SLOPE 0.2f

typedef float v8f __attribute__((ext_vector_type(8)));
typedef float v2f __attribute__((ext_vector_type(2)));

// ---------- monotone float<->uint mapping for atomic segment-max ----------
__device__ __forceinline__ unsigned int enc_f32(float x) {
    unsigned int b = __float_as_uint(x);
    return (b & 0x80000000u) ? ~b : (b | 0x80000000u);
}
__device__ __forceinline__ float dec_f32(unsigned int u) {
    unsigned int b = (u & 0x80000000u) ? (u ^ 0x80000000u) : ~u;
    return __uint_as_float(b);
}

// ---------- K0: init out = bias, mx = 0 (encoded sentinel), sm = 0 ----------
__global__ void k_init(float* __restrict__ out, const float* __restrict__ bias,
                       unsigned int* __restrict__ mx, float* __restrict__ sm) {
    int i = blockIdx.x * blockDim.x + threadIdx.x;
    if (i < N_NODES * HD) out[i] = bias[i & 255];
    if (i < N_NODES * NUM_HEADS) { mx[i] = 0u; sm[i] = 0.0f; }
}

// ---------- Kq: q[k*8+h] = sum_d W_e[k, h*32+d] * attn_e[h*32+d] ----------
__global__ void k_q(const float* __restrict__ W_e, const float* __restrict__ attn_e,
                    float* __restrict__ q) {
    int t = threadIdx.x;
    if (t < 64) {
        int k = t >> 3, hh = t & 7;
        float s = 0.0f;
        #pragma unroll
        for (int d = 0; d < OUT_FEATS; ++d)
            s += W_e[k * HD + hh * OUT_FEATS + d] * attn_e[hh * OUT_FEATS + d];
        q[t] = s;
    }
}

// ---------- K1: h = feat @ W_src via V_WMMA_F32_16X16X4_F32 ----------
// Block: 256 threads (8 waves). Block computes 16 rows x 256 cols.
// Wave w: cols [w*32, w*32+32) -> two 16x16 output tiles (acc0, acc1).
__global__ void __launch_bounds__(256) k_gemm(const float* __restrict__ feat,
                                              const float* __restrict__ W,
                                              float* __restrict__ h) {
    const int lane  = threadIdx.x & 31;
    const int wave  = threadIdx.x >> 5;
    const int rowBase = blockIdx.x * 16;

    const int arow  = lane & 15;        // A: lanes 0-15 = rows, replicated in 16-31
    const int khalf = lane >> 4;        // 0: K{0,1}; 1: K{2,3}
    const int bcol0 = wave * 32 + (lane & 15);
    const int bcol1 = bcol0 + 16;

    const float* Arow = feat + (size_t)(rowBase + arow) * IN_FEATS;

    v8f acc0 = {}; v8f acc1 = {};

    #pragma unroll 4
    for (int k = 0; k < IN_FEATS; k += 4) {
        // A fragment (16x4): v0/v1 hold K = k + 2*khalf + {0,1}
        v2f a = *(const v2f*)(Arow + k + 2 * khalf);
        // B fragments (4x16): v0 = row (k+2*khalf), v1 = row (k+2*khalf+1), col across lanes
        int kb = k + 2 * khalf;
        v2f b0, b1;
        b0.x = W[(size_t)(kb + 0) * HD + bcol0];
        b0.y = W[(size_t)(kb + 1) * HD + bcol0];
        b1.x = W[(size_t)(kb + 0) * HD + bcol1];
        b1.y = W[(size_t)(kb + 1) * HD + bcol1];
        acc0 = __builtin_amdgcn_wmma_f32_16x16x4_f32(false, a, false, b0, (short)0, acc0, false, false);
        acc1 = __builtin_amdgcn_wmma_f32_16x16x4_f32(false, a, false, b1, (short)0, acc1, false, false);
    }

    // C/D layout: vgpr g -> row g (lanes 0-15) / row 8+g (lanes 16-31), col = lane&15
    const int srow = rowBase + (lane >> 4) * 8;
    const int scol = lane & 15;
    #pragma unroll
    for (int g = 0; g < 8; ++g) {
        h[(size_t)(srow + g) * HD + wave * 32 + scol]      = acc0[g];
        h[(size_t)(srow + g) * HD + wave * 32 + 16 + scol] = acc1[g];
    }
}

// ---------- K2: el/er per node (one wave per node, shuffle reduce) ----------
__global__ void k_node_attn(const float* __restrict__ h,
                            const float* __restrict__ attn_l,
                            const float* __restrict__ attn_r,
                            float* __restrict__ el, float* __restrict__ er) {
    int node = blockIdx.x * (blockDim.x >> 5) + (threadIdx.x >> 5);
    int lane = threadIdx.x & 31;
    if (node >= N_NODES) return;
    int head  = lane >> 2;            // 4 lanes per head
    int dbase = (lane & 3) * 8;       // 8 elems per lane, stays within one head
    const float* hp = h + (size_t)node * HD + head * OUT_FEATS + dbase;
    const float* al = attn_l + head * OUT_FEATS + dbase;
    const float* ar = attn_r + head * OUT_FEATS + dbase;
    float sl = 0.0f, sr = 0.0f;
    #pragma unroll
    for (int i = 0; i < 8; ++i) { float v = hp[i]; sl += v * al[i]; sr += v * ar[i]; }
    sl += __shfl_xor(sl, 1); sl += __shfl_xor(sl, 2);
    sr += __shfl_xor(sr, 1); sr += __shfl_xor(sr, 2);
    if ((lane & 3) == 0) {
        el[node * NUM_HEADS + head] = sl;
        er[node * NUM_HEADS + head] = sr;
    }
}

// ---------- K3: logits + segment max over dst ----------
__global__ void k_edge_logits(const float* __restrict__ edge_emb,
                              const int* __restrict__ src, const int* __restrict__ dst,
                              const float* __restrict__ el, const float* __restrict__ er,
                              const float* __restrict__ q,
                              float* __restrict__ logits, unsigned int* __restrict__ mx) {
    int t = blockIdx.x * blockDim.x + threadIdx.x;
    if (t >= N_EDGES * NUM_HEADS) return;
    int e = t >> 3, hh = t & 7;
    int s = src[e], d = dst[e];
    float ee = 0.0f;
    #pragma unroll
    for (int k = 0; k < NUM_HEADS; ++k)
        ee += edge_emb[(size_t)e * NUM_HEADS + k] * q[k * NUM_HEADS + hh];
    float x = el[s * NUM_HEADS + hh] + er[d * NUM_HEADS + hh] + ee;
    x = (x > 0.0f) ? x : NEG_SLOPE * x;   // leaky_relu
    logits[t] = x;
    atomicMax(&mx[d * NUM_HEADS + hh], enc_f32(x));
}

// ---------- K4: p = exp(logit - m[dst]); segment sum ----------
__global__ void k_edge_exp(const int* __restrict__ dst,
                           const unsigned int* __restrict__ mx,
                           float* __restrict__ logits, float* __restrict__ sm) {
    int t = blockIdx.x * blockDim.x + threadIdx.x;
    if (t >= N_EDGES * NUM_HEADS) return;
    int e = t >> 3, hh = t & 7;
    int d = dst[e];
    unsigned int em = mx[d * NUM_HEADS + hh];
    float m = (em == 0u) ? 0.0f : dec_f32(em);   // sentinel -> 0 (matches isfinite fixup)
    float p = __expf(logits[t] - m);
    logits[t] = p;                                // reuse buffer to hold p
    atomicAdd(&sm[d * NUM_HEADS + hh], p);
}

// ---------- K5: alpha-weighted scatter: out[dst] += alpha * h[src] ----------
__global__ void k_edge_scatter(const int* __restrict__ src, const int* __restrict__ dst,
                               const float* __restrict__ p, const float* __restrict__ sm,
                               const float* __restrict__ h, float* __restrict__ out) {
    int t = blockIdx.x * blockDim.x + threadIdx.x;
    if (t >= N_EDGES * NUM_HEADS) return;
    int e = t >> 3, hh = t & 7;
    int s = src[e], d = dst[e];
    float alpha = p[t] / (sm[d * NUM_HEADS + hh] + 1e-9f);
    const float* hp = h + (size_t)s * HD + hh * OUT_FEATS;
    float* op = out + (size_t)d * HD + hh * OUT_FEATS;
    #pragma unroll
    for (int i = 0; i < OUT_FEATS; i += 4) {
        float4 v = *(const float4*)(hp + i);
        atomicAdd(op + i + 0, alpha * v.x);
        atomicAdd(op + i + 1, alpha * v.y);
        atomicAdd(op + i + 2, alpha * v.z);
        atomicAdd(op + i + 3, alpha * v.w);
    }
}

extern "C" void kernel_launch(void* const* d_in, const int* in_sizes, int n_in,
                              void* d_out, int out_size, void* d_ws, size_t ws_size,
                              hipStream_t stream) {
    const float* feat     = (const float*)d_in[0];
    const float* edge_emb = (const float*)d_in[1];
    const int*   src      = (const int*)d_in[2];
    const int*   dst      = (const int*)d_in[3];
    const float* W_src    = (const float*)d_in[4];
    const float* W_e      = (const float*)d_in[5];
    const float* attn_l   = (const float*)d_in[6];
    const float* attn_r   = (const float*)d_in[7];
    const float* attn_e   = (const float*)d_in[8];
    const float* bias     = (const float*)d_in[9];
    float* out = (float*)d_out;

    // workspace layout (floats)
    float* ws = (float*)d_ws;
    float*        h_buf  = ws;                                   // N*256
    float*        el     = h_buf  + (size_t)N_NODES * HD;        // N*8
    float*        er     = el     + (size_t)N_NODES * NUM_HEADS; // N*8
    unsigned int* mx     = (unsigned int*)(er + (size_t)N_NODES * NUM_HEADS); // N*8
    float*        sm     = (float*)(mx + (size_t)N_NODES * NUM_HEADS);        // N*8
    float*        q      = sm + (size_t)N_NODES * NUM_HEADS;     // 64
    float*        logits = q + 64;                               // E*8

    // K0: init out/mx/sm
    k_init<<<(N_NODES * HD + 255) / 256, 256, 0, stream>>>(out, bias, mx, sm);
    // Kq: 8x8 folded edge-attention matrix
    k_q<<<1, 64, 0, stream>>>(W_e, attn_e, q);
    // K1: WMMA GEMM h = feat @ W_src
    k_gemm<<<N_NODES / 16, 256, 0, stream>>>(feat, W_src, h_buf);
    // K2: el / er
    k_node_attn<<<(N_NODES + 7) / 8, 256, 0, stream>>>(h_buf, attn_l, attn_r, el, er);
    // K3: edge logits + segment max
    int ethreads = N_EDGES * NUM_HEADS;
    k_edge_logits<<<(ethreads + 255) / 256, 256, 0, stream>>>(edge_emb, src, dst, el, er, q, logits, mx);
    // K4: exp + segment sum
    k_edge_exp<<<(ethreads + 255) / 256, 256, 0, stream>>>(dst, mx, logits, sm);
    // K5: weighted scatter-add
    k_edge_scatter<<<(ethreads + 255) / 256, 256, 0, stream>>>(src, dst, logits, sm, h_buf, out);
}